// NearestScatterRasterizer_82368882802956
// MI455X (gfx1250) — compile-verified
//
#include <hip/hip_runtime.h>
#include <stdint.h>

typedef __attribute__((ext_vector_type(2))) float v2f;
typedef __attribute__((ext_vector_type(8))) float v8f;

#define BLK 256
#define DD  8   // DESC_DIM

// ---------------------------------------------------------------------------
// Pass 0: init packed z/index buffer to "empty" (all ones = +inf key)
// ---------------------------------------------------------------------------
__global__ void zbuf_init_kernel(unsigned long long* __restrict__ zbuf, long long count) {
  long long i = (long long)blockIdx.x * blockDim.x + threadIdx.x;
  if (i < count) zbuf[i] = 0xFFFFFFFFFFFFFFFFull;
}

// ---------------------------------------------------------------------------
// Pass 1: project points with V_WMMA_F32_16X16X4_F32 and scatter packed
//         (z_bits<<32)|idx keys with a single u64 atomic-min per point.
//
// Per wave32: two WMMAs cover 32 points.
//   A (16x4 f32, 2 VGPRs): lanes 0-15 hold K=0 (V0) / K=1 (V1) of row M=lane,
//                          lanes 16-31 hold K=2 / K=3 of row M=lane-16.
//     -> lane provides [x,y] (half 0) or [z,1] (half 1) of point base+m.
//   B (4x16 f32, 2 VGPRs): lanes 0-15 hold rows K=0 (V0) / K=1 (V1) at N=lane,
//                          lanes 16-31 hold rows K=2 / K=3 at N=lane-16.
//     -> B rows 0..2 = R rows, row 3 = T; columns >= 3 are zero.
//   D (16x16 f32, 8 VGPRs): VGPR r: lanes 0-15 -> (M=r,   N=lane),
//                                    lanes16-31 -> (M=r+8, N=lane-16).
//     -> columns N=0,1,2 are cam x/y/z; staged through LDS to the owning lane.
// ---------------------------------------------------------------------------
__global__ __launch_bounds__(BLK) void project_scatter_kernel(
    const float* __restrict__ points,      // [b][n][3]
    const float* __restrict__ R,           // [b][3][3]
    const float* __restrict__ T,           // [b][3]
    const float* __restrict__ fcl,         // [b][2]
    const float* __restrict__ prp,         // [b][2]
    unsigned long long* __restrict__ zbuf, // [b][H*W]
    int n, int H, int W, float sS, float cxc, float cyc)
{
  const int batch = blockIdx.y;
  const int lane  = threadIdx.x & 31;
  const int wave  = threadIdx.x >> 5;
  const int m     = lane & 15;   // column / row-in-half index
  const int hf    = lane >> 4;   // which K-half this lane supplies

  const float* Rb = R + batch * 9;
  const float* Tb = T + batch * 3;
  const float f0 = fcl[batch * 2 + 0], f1 = fcl[batch * 2 + 1];
  const float p0 = prp[batch * 2 + 0], p1 = prp[batch * 2 + 1];

  const int g0 = blockIdx.x * BLK + wave * 32;   // first point of this wave

  // Clamp tail loads so EXEC stays all-ones through the WMMAs.
  int q0 = g0 + m;       if (q0 > n - 1) q0 = n - 1;
  int q1 = g0 + 16 + m;  if (q1 > n - 1) q1 = n - 1;
  const float* P = points + (size_t)batch * n * 3;
  float x0 = P[(size_t)q0 * 3 + 0], y0 = P[(size_t)q0 * 3 + 1], z0 = P[(size_t)q0 * 3 + 2];
  float x1 = P[(size_t)q1 * 3 + 0], y1 = P[(size_t)q1 * 3 + 1], z1 = P[(size_t)q1 * 3 + 2];

  v2f a0, a1, bf;
  a0.x = hf ? z0 : x0;  a0.y = hf ? 1.0f : y0;
  a1.x = hf ? z1 : x1;  a1.y = hf ? 1.0f : y1;

  float b0 = 0.0f, b1 = 0.0f;
  if (m < 3) {
    b0 = hf ? Rb[2 * 3 + m] : Rb[0 * 3 + m];  // V0: K=2 | K=0
    b1 = hf ? Tb[m]         : Rb[1 * 3 + m];  // V1: K=3 | K=1
  }
  bf.x = b0; bf.y = b1;

  v8f c0 = {};
  v8f d0 = __builtin_amdgcn_wmma_f32_16x16x4_f32(false, a0, false, bf, (short)0, c0, false, false);
  v8f d1 = __builtin_amdgcn_wmma_f32_16x16x4_f32(false, a1, false, bf, (short)0, c0, false, false);

  // Redistribute D columns 0..2 (cam x/y/z) so each lane owns its point.
  __shared__ float camS[BLK][3];
  const int rb = hf ? 8 : 0;
  if (m < 3) {
#pragma unroll
    for (int r = 0; r < 8; ++r) {
      camS[wave * 32 + rb + r][m]      = d0[r];  // points g0 +  0..15
      camS[wave * 32 + 16 + rb + r][m] = d1[r];  // points g0 + 16..31
    }
  }
  __syncthreads();

  const float camx = camS[threadIdx.x][0];
  const float camy = camS[threadIdx.x][1];
  const float camz = camS[threadIdx.x][2];

  const int pt = blockIdx.x * BLK + threadIdx.x;  // point index within batch
  if (pt < n && camz > 0.0f) {
    float ndcx = f0 * camx / camz + p0;
    float ndcy = f1 * camy / camz + p1;
    float sx = -(sS * ndcx + cxc);
    float sy = -(sS * ndcy + cyc);
    int ix = (int)(sx - 1e-6f + 0.5f);   // trunc-toward-zero, matches astype(int32)
    int iy = (int)(sy - 1e-6f + 0.5f);
    if (ix >= 0 && ix < W && iy >= 0 && iy < H) {
      // z > 0 => float bits are monotonic as u32; tie-break = smaller index.
      unsigned long long key =
          ((unsigned long long)__float_as_uint(camz) << 32) | (unsigned int)pt;
      atomicMin(zbuf + (size_t)batch * H * W + (size_t)iy * W + ix, key);
    }
  }
}

// ---------------------------------------------------------------------------
// Pass 2: per output pixel, read the 4 supersample winners, gather their
// descriptors, average, normalize by coverage, background-fill.
//   feat = sum/cnt  (== (sum/4)/(cnt/4) bit-exactly: /4 is exact),
//   mask = cnt ? 1 : 0.
// ---------------------------------------------------------------------------
__global__ void resolve_kernel(
    const unsigned long long* __restrict__ zbuf, // [b][H*W]
    const float* __restrict__ desc,              // [b][n][8]
    const float* __restrict__ bg,                // [8]
    float* __restrict__ out,                     // [b][9][Ho][Wo]
    int n, int Ho, int Wo, long long total)
{
  long long idx = (long long)blockIdx.x * blockDim.x + threadIdx.x;
  if (idx >= total) return;
  const long long hw = (long long)Ho * Wo;
  const int batch = (int)(idx / hw);
  const int rem   = (int)(idx - (long long)batch * hw);
  const int y = rem / Wo, x = rem % Wo;
  const int H = 2 * Ho, W = 2 * Wo;

  const unsigned long long* zb = zbuf + (size_t)batch * H * W;
  float acc[DD];
#pragma unroll
  for (int c = 0; c < DD; ++c) acc[c] = 0.0f;
  int cnt = 0;

#pragma unroll
  for (int dy = 0; dy < 2; ++dy) {
#pragma unroll
    for (int dx = 0; dx < 2; ++dx) {
      unsigned long long e = zb[(size_t)(2 * y + dy) * W + (2 * x + dx)];
      if (e != 0xFFFFFFFFFFFFFFFFull) {
        unsigned pi = (unsigned)e;  // winning point index
        const float4* dp = (const float4*)(desc + ((size_t)batch * n + pi) * DD);
        float4 da = dp[0], db = dp[1];
        acc[0] += da.x; acc[1] += da.y; acc[2] += da.z; acc[3] += da.w;
        acc[4] += db.x; acc[5] += db.y; acc[6] += db.z; acc[7] += db.w;
        ++cnt;
      }
    }
  }

  const size_t cs = (size_t)Ho * Wo;                       // channel stride
  float* ob = out + (size_t)batch * (DD + 1) * cs + (size_t)y * Wo + x;
  if (cnt) {
    const float fc = (float)cnt;
#pragma unroll
    for (int c = 0; c < DD; ++c) ob[c * cs] = acc[c] / fc;
    ob[DD * cs] = 1.0f;
  } else {
#pragma unroll
    for (int c = 0; c < DD; ++c) ob[c * cs] = bg[c];
    ob[DD * cs] = 0.0f;
  }
}

// ---------------------------------------------------------------------------
extern "C" void kernel_launch(void* const* d_in, const int* in_sizes, int n_in,
                              void* d_out, int out_size, void* d_ws, size_t ws_size,
                              hipStream_t stream) {
  const float* points = (const float*)d_in[0];
  const float* desc   = (const float*)d_in[1];
  const float* R      = (const float*)d_in[2];
  const float* T      = (const float*)d_in[3];
  const float* fcl    = (const float*)d_in[4];
  const float* prp    = (const float*)d_in[5];
  const float* bg     = (const float*)d_in[6];
  float* out = (float*)d_out;

  const int b = in_sizes[2] / 9;                       // R_row: b*3*3
  if (b <= 0) return;
  const long long n = (long long)in_sizes[0] / (3LL * b);
  if (n <= 0) return;
  const long long hw = (long long)out_size / ((long long)b * (DD + 1)); // Ho*Wo
  int Wo = 1; while ((long long)Wo * Wo < hw) ++Wo;    // square output
  const int Ho = (int)(hw / Wo);
  const int H = 2 * Ho, W = 2 * Wo;                    // SS = 2
  const long long HW = (long long)H * W;

  unsigned long long* zbuf = (unsigned long long*)d_ws;
  if (ws_size < (size_t)b * HW * 8ull) return;         // need b*H*W u64 keys

  const long long zcount = (long long)b * HW;
  zbuf_init_kernel<<<(unsigned)((zcount + 511) / 512), 512, 0, stream>>>(zbuf, zcount);

  const float sS  = (float)((((H < W) ? H : W) - 1) * 0.5);
  const float cxc = (float)(-(W - 1) * 0.5);
  const float cyc = (float)(-(H - 1) * 0.5);

  dim3 grid((unsigned)((n + BLK - 1) / BLK), (unsigned)b, 1);
  project_scatter_kernel<<<grid, BLK, 0, stream>>>(points, R, T, fcl, prp, zbuf,
                                                   (int)n, H, W, sS, cxc, cyc);

  const long long total = (long long)b * Ho * Wo;
  resolve_kernel<<<(unsigned)((total + 255) / 256), 256, 0, stream>>>(
      zbuf, desc, bg, out, (int)n, Ho, Wo, total);
}